// crossMSEloss_31937376813111
// MI455X (gfx1250) — compile-verified
//
#include <hip/hip_runtime.h>

typedef __attribute__((ext_vector_type(2))) float v2f;
typedef __attribute__((ext_vector_type(8))) float v8f;

#define PP   64   // P (points) -> both GEMM dims
#define DD   63   // D
#define KPAD 64   // K padded to 64 (16 steps of K=4)

// One workgroup per (b,s) slice. 512 threads = 16 wave32s.
// dist^2 = ||a||^2 + ||b||^2 - 2 a.b; cross term via V_WMMA_F32_16X16X4_F32.
// min/sqrt commute (sqrt monotonic), so all mins are taken on d^2 in registers;
// sqrt applied only to the 128 final per-row/per-col minima.
__launch_bounds__(512, 2)
__global__ void crossmse_wmma_kernel(const float* __restrict__ inputs,
                                     const float* __restrict__ targets,
                                     float* __restrict__ out)
{
    __shared__ float ldsA  [PP * KPAD];   // preds, col 63 zeroed (K pad)
    __shared__ float ldsBT [KPAD * PP];   // targets^T (K-major), row 63 zeroed
    __shared__ float npartP[4 * PP];      // partial ||preds[p]||^2
    __shared__ float npartT[4 * PP];      // partial ||targets[t]||^2
    __shared__ float pn[PP], tn[PP];
    __shared__ float conf[PP], invConf2[PP];
    __shared__ float rowPart[PP * 4];     // min_t d^2 per (row p, ni)
    __shared__ float colPart[PP * 4];     // min_p d^2*invConf^2 per (col t, mi)
    __shared__ float partial[2 * PP];

    const int bs  = blockIdx.x;           // 0 .. B*S-1 (1024)
    const int tid = threadIdx.x;          // 0 .. 511

    const float* inBase = inputs  + (size_t)bs * (PP * (DD + 1)); // 64*64 fp32, 16B aligned
    const float* tgBase = targets + (size_t)bs * (PP * DD);       // 64*63 fp32, 16B aligned

    // ---- Stage preds (+conf): 4096 floats; 2x b128 loads + 2x b128 LDS stores/thread ----
    {
        const int p  = tid >> 3;              // row (constant per thread)
        const int c0 = (tid & 7) * 8;         // first col of this thread's 8
        float4 v0 = *(const float4*)(inBase + p * 64 + c0);
        float4 v1 = *(const float4*)(inBase + p * 64 + c0 + 4);
        if (c0 + 7 == DD) {                   // last quad of the row holds conf at k=63
            conf[p] = v1.w;
            v1.w = 0.0f;                      // zero the K-pad column
        }
        *(float4*)(ldsA + p * KPAD + c0)     = v0;
        *(float4*)(ldsA + p * KPAD + c0 + 4) = v1;
    }
    // ---- Stage targets transposed (K-major): 4032 floats, scalar scatter ----
    #pragma unroll
    for (int i = 0; i < 8; ++i) {
        int e = tid + i * 512;
        if (e < PP * DD) {
            int t = e / DD, k = e - t * DD;
            ldsBT[k * PP + t] = tgBase[e];
        }
    }
    if (tid < PP) ldsBT[DD * PP + tid] = 0.0f;   // zero K-pad row
    __syncthreads();

    // ---- Gram matrix G = preds @ targets^T via FP32 WMMA 16x16x4 ----
    const int wave = tid >> 5;
    const int lane = tid & 31;
    const int half = lane >> 4;      // 0: lanes 0-15, 1: lanes 16-31
    const int lm   = lane & 15;
    const int mi   = wave >> 2, ni = wave & 3;

    v8f acc = {};
    #pragma unroll
    for (int kk = 0; kk < 16; ++kk) {
        const int k0 = kk * 4 + 2 * half;
        v2f a, b;
        a.x = ldsA[(mi * 16 + lm) * KPAD + k0];
        a.y = ldsA[(mi * 16 + lm) * KPAD + k0 + 1];
        b.x = ldsBT[(k0    ) * PP + ni * 16 + lm];
        b.y = ldsBT[(k0 + 1) * PP + ni * 16 + lm];
        acc = __builtin_amdgcn_wmma_f32_16x16x4_f32(
                  false, a, false, b, (short)0, acc, false, false);
    }

    // ---- Parallel row norms: 512 threads x 16-element partial sums ----
    {
        const int idx = tid & 63;        // p or t
        const int q   = (tid >> 6) & 3;  // k quarter
        if (tid < 256) {
            float s = 0.0f;
            #pragma unroll
            for (int k = 0; k < 16; ++k) {
                float v = ldsA[idx * KPAD + q * 16 + k];  // k=63 entry is 0
                s += v * v;
            }
            npartP[q * PP + idx] = s;
        } else {
            float s = 0.0f;
            #pragma unroll
            for (int k = 0; k < 16; ++k) {
                float v = ldsBT[(q * 16 + k) * PP + idx]; // k=63 row is 0
                s += v * v;
            }
            npartT[q * PP + idx] = s;
        }
    }
    __syncthreads();
    if (tid < PP) {
        pn[tid] = npartP[tid] + npartP[PP + tid] + npartP[2 * PP + tid] + npartP[3 * PP + tid];
        float c = conf[tid];
        invConf2[tid] = 1.0f / (c * c);
    } else if (tid < 2 * PP) {
        int t = tid - PP;
        tn[t] = npartT[t] + npartT[PP + t] + npartT[2 * PP + t] + npartT[3 * PP + t];
    }
    __syncthreads();

    // ---- d^2 + in-register reductions straight from the WMMA accumulator ----
    {
        const int t    = ni * 16 + lm;
        const float tnt = tn[t];
        float cmin = 3.4e38f;                 // min over this tile's 16 rows (scaled)
        #pragma unroll
        for (int j = 0; j < 8; ++j) {
            const int p = mi * 16 + j + 8 * half;   // C/D layout: M = j + 8*half, N = lm
            float d2 = fmaxf(pn[p] + tnt - 2.0f * acc[j], 0.0f);
            cmin = fminf(cmin, d2 * invConf2[p]);
            // row min over the 16 columns held across this half-wave's lanes
            float r = d2;
            r = fminf(r, __shfl_xor(r, 1));
            r = fminf(r, __shfl_xor(r, 2));
            r = fminf(r, __shfl_xor(r, 4));
            r = fminf(r, __shfl_xor(r, 8));
            if (lm == 0) rowPart[p * 4 + ni] = r;
        }
        cmin = fminf(cmin, __shfl_xor(cmin, 16));   // combine halves (rows 0-7 vs 8-15)
        if (half == 0) colPart[t * 4 + mi] = cmin;
    }
    __syncthreads();

    // ---- final: 64 row mins -> sqrt*conf ; 64 col mins -> sqrt ----
    if (tid < PP) {
        const int p = tid;
        float m = fminf(fminf(rowPart[p * 4 + 0], rowPart[p * 4 + 1]),
                        fminf(rowPart[p * 4 + 2], rowPart[p * 4 + 3]));
        partial[p] = conf[p] * __builtin_sqrtf(m);
    } else if (tid < 2 * PP) {
        const int t = tid - PP;
        float m = fminf(fminf(colPart[t * 4 + 0], colPart[t * 4 + 1]),
                        fminf(colPart[t * 4 + 2], colPart[t * 4 + 3]));
        partial[PP + t] = __builtin_sqrtf(m);
    }
    __syncthreads();

    // ---- wave32 sum of 128 partials ----
    if (tid < 32) {
        float s = partial[tid] + partial[tid + 32] + partial[tid + 64] + partial[tid + 96];
        #pragma unroll
        for (int off = 16; off > 0; off >>= 1)
            s += __shfl_down(s, off, 32);
        if (tid == 0) out[bs] = s;
    }
}

extern "C" void kernel_launch(void* const* d_in, const int* in_sizes, int n_in,
                              void* d_out, int out_size, void* d_ws, size_t ws_size,
                              hipStream_t stream) {
    (void)in_sizes; (void)n_in; (void)d_ws; (void)ws_size; (void)out_size;
    const float* inputs  = (const float*)d_in[0];  // (4,256,64,64) fp32
    const float* targets = (const float*)d_in[1];  // (4,256,64,63) fp32
    float* out = (float*)d_out;                    // (4,256) fp32
    crossmse_wmma_kernel<<<dim3(4 * 256), dim3(512), 0, stream>>>(inputs, targets, out);
}